// EnhancedLoss_61890478735514
// MI455X (gfx1250) — compile-verified
//
#include <hip/hip_runtime.h>
#include <hip/hip_bf16.h>
#include <stdint.h>

// ---------------- constants ----------------
#define NCLS 80
#define T0C 0.0025f
#define T1C 0.0225f

typedef unsigned int u32;
typedef u32   u32x4 __attribute__((ext_vector_type(4)));
typedef int   i32x4 __attribute__((ext_vector_type(4)));
typedef int   i32x8 __attribute__((ext_vector_type(8)));
typedef float f32x2 __attribute__((ext_vector_type(2)));
typedef float f32x8 __attribute__((ext_vector_type(8)));

// workspace layout (32-bit units)
//  lists:   48 lists * 512 entries * 10 u32   (cell, obj, w, box[4], mask[3])
//  counts:  48 u32
//  results: 24576 entries * 4 f32             (ocorr, box_per, cls_per, is_pos)
//  partials: obj softplus partial sums per k_obj block
#define LISTS_U   0u
#define COUNTS_U  (48u * 512u * 10u)           // 245760
#define RES_U     (COUNTS_U + 48u)             // 245808
#define PART_U    (RES_U + 24576u * 4u)        // 344112

#define NBLK0 300   // 307200 / 1024
#define NBLK1 75    //  76800 / 1024
#define NBLK2 19    //  ceil(19200/1024)
#define NPART (NBLK0 + NBLK1 + NBLK2)

__device__ __forceinline__ float softplusf(float x) {
    // logaddexp(0, x)
    return fmaxf(x, 0.f) + log1pf(expf(-fabsf(x)));
}

// ---------------- kernel 1: order-faithful target scatter -> compact lists ----
__device__ __forceinline__ int find_or_add(u32* ent, int& count, u32 cell) {
    for (int i = 0; i < count; ++i)
        if (ent[i * 10] == cell) return i;
    int i = (count < 512) ? count++ : 511;
    u32* e = ent + i * 10;
    float* f = (float*)e;
    e[0] = cell;
    f[1] = 0.f;   // obj default
    f[2] = 1.f;   // weight map default is ones
    f[3] = f[4] = f[5] = f[6] = 0.f;
    e[7] = e[8] = e[9] = 0u;
    return i;
}

__global__ __launch_bounds__(64)
void k_build(const float* __restrict__ tgt, u32* __restrict__ ws) {
    __shared__ u32 ent[512 * 10];
    __shared__ int cnt_sh;
    const int list = blockIdx.x;           // list = s*16 + b
    const int s = list >> 4;
    const int b = list & 15;
    const int G = (s == 0) ? 80 : ((s == 1) ? 40 : 20);

    if (threadIdx.x == 0) {
        int count = 0;
        for (int t = 0; t < 100; ++t) {
            const float* T = tgt + (size_t)(b * 100 + t) * 5;
            float x1 = fminf(fmaxf(T[0], 0.f), 1.f);
            float y1 = fminf(fmaxf(T[1], 0.f), 1.f);
            float x2 = fminf(fmaxf(T[2], 0.f), 1.f);
            float y2 = fminf(fmaxf(T[3], 0.f), 1.f);
            bool valid = (x2 > x1) && (y2 > y1);
            float area = (x2 - x1) * (y2 - y1);
            int sidx = (area <= T0C) ? 0 : ((area <= T1C) ? 1 : 2);
            valid = valid && (sidx == s);
            if (!valid) continue;          // neighbors also require 'valid'
            int gi = (int)floorf(0.5f * (x1 + x2) * (float)G);
            int gj = (int)floorf(0.5f * (y1 + y2) * (float)G);
            gi = min(max(gi, 0), G - 1);
            gj = min(max(gj, 0), G - 1);
            int cid = (int)T[4];
            bool cls_ok = (cid >= 0) && (cid < NCLS);
            float smallness = 1.f - fminf(1.f, area / T0C);
            float weight = 1.f + fmaxf(0.f, smallness);   // BOOST = 2

            // center cell: full overwrite (obj=1, box=bvec, w=weight), cls bit OR'd
            {
                int idx = find_or_add(ent, count, (u32)(gj * G + gi));
                u32* e = ent + idx * 10; float* f = (float*)e;
                f[1] = 1.f; f[2] = weight;
                f[3] = x1; f[4] = y1; f[5] = x2; f[6] = y2;
                if (cls_ok) e[7 + (cid >> 5)] |= (1u << (cid & 31));
            }
            // tiny targets spread to 4-neighbors: obj=max(.,0.5), box overwritten,
            // w=max(., 0.5*weight), cls bit OR'd
            if (area <= T0C) {
                const int dj[4] = {-1, 1, 0, 0};
                const int di[4] = {0, 0, -1, 1};
                for (int n = 0; n < 4; ++n) {
                    int nj = gj + dj[n], ni = gi + di[n];
                    if (nj < 0 || nj >= G || ni < 0 || ni >= G) continue;
                    int idx = find_or_add(ent, count, (u32)(nj * G + ni));
                    u32* e = ent + idx * 10; float* f = (float*)e;
                    f[1] = fmaxf(f[1], 0.5f);
                    f[2] = fmaxf(f[2], 0.5f * weight);
                    f[3] = x1; f[4] = y1; f[5] = x2; f[6] = y2;
                    if (cls_ok) e[7 + (cid >> 5)] |= (1u << (cid & 31));
                }
            }
        }
        cnt_sh = count;
        ws[COUNTS_U + list] = (u32)count;
    }
    __syncthreads();
    int count = cnt_sh;
    for (int i = threadIdx.x; i < count * 10; i += blockDim.x)
        ws[LISTS_U + (u32)list * 5120u + (u32)i] = ent[i];
}

// ---------------- kernel 2: obj-channel softplus sum via TDM + WMMA ----------
__global__ __launch_bounds__(256)
void k_obj(const float* __restrict__ o0, const float* __restrict__ o1,
           const float* __restrict__ o2, u32* __restrict__ ws) {
    __shared__ float tile[1024];
    __shared__ float wsum[8];
    const int bid = blockIdx.x;
    const float* base; u32 row0, R;
    if (bid < NBLK0)               { base = o0; row0 = (u32)bid * 1024u;                 R = 307200u; }
    else if (bid < NBLK0 + NBLK1)  { base = o1; row0 = (u32)(bid - NBLK0) * 1024u;       R = 76800u;  }
    else                           { base = o2; row0 = (u32)(bid - NBLK0 - NBLK1) * 1024u; R = 19200u; }
    const u32 rem  = R - row0;                     // remaining valid rows (tensor_dim1)
    const u32 vlim = rem < 1024u ? rem : 1024u;

    if (threadIdx.x < 32) {
        // D#: 2D tensor, data_size=4B, dim0=85 floats/row, stride 85,
        // tile = 1 x 1024 -> strided gather of the obj logit into LDS.
        uint64_t ga = (uint64_t)(uintptr_t)base + ((uint64_t)row0 * 85ull + 4ull) * 4ull;
        u32 ldsOff = (u32)(uintptr_t)(&tile[0]);   // LDS addr = generic addr[31:0]
        u32x4 g0 = { 1u,                                            // count=1
                     ldsOff,
                     (u32)(ga & 0xFFFFFFFFull),
                     (u32)((ga >> 32) & 0x1FFFFFFull) | (2u << 30) };// type=2 (image)
        i32x8 g1 = { (int)(2u << 16),                               // data_size = 4B
                     (int)(85u << 16),                              // tensor_dim0 lo16
                     (int)((rem & 0xFFFFu) << 16),                  // dim0 hi=0 | dim1 lo16
                     (int)(((rem >> 16) & 0xFFFFu) | (1u << 16)),   // dim1 hi | tile_dim0=1
                     (int)1024,                                     // tile_dim1=1024, tile_dim2=0
                     85,                                            // tensor_dim0_stride lo32
                     0, 0 };
        i32x4 z4 = {0, 0, 0, 0};
#if defined(__clang_major__) && __clang_major__ >= 23
        i32x8 z8 = {0, 0, 0, 0, 0, 0, 0, 0};
        __builtin_amdgcn_tensor_load_to_lds(g0, g1, z4, z4, z8, 0);
#else
        __builtin_amdgcn_tensor_load_to_lds(g0, g1, z4, z4, 0);
#endif
        __builtin_amdgcn_s_wait_tensorcnt(0);
    }
    __syncthreads();

    // WMMA-as-reduction: D = A x ones(4x16) + C sums 64 floats per issue.
    const int lane = threadIdx.x & 31;
    const int wv   = threadIdx.x >> 5;
    f32x8 acc = {0.f, 0.f, 0.f, 0.f, 0.f, 0.f, 0.f, 0.f};
    f32x2 onesv = {1.f, 1.f};
    for (int k = 0; k < 2; ++k) {
        u32 i0 = (u32)wv * 128u + (u32)k * 64u + (u32)lane * 2u;
        f32x2 a;
        a.x = (i0     < vlim) ? softplusf(tile[i0])     : 0.f;
        a.y = (i0 + 1 < vlim) ? softplusf(tile[i0 + 1]) : 0.f;
        acc = __builtin_amdgcn_wmma_f32_16x16x4_f32(
                  false, a, false, onesv, (short)0, acc, false, false);
    }
    float sum = acc[0] + acc[1] + acc[2] + acc[3] + acc[4] + acc[5] + acc[6] + acc[7];
    sum += __shfl_down(sum, 16, 32);     // lane0 += lane16 half
    if (lane == 0) wsum[wv] = sum;
    __syncthreads();
    if (threadIdx.x == 0) {
        float t = 0.f;
        for (int i = 0; i < 8; ++i) t += wsum[i];
        ((float*)ws)[PART_U + (u32)bid] = t;
    }
}

// ---------------- kernel 3: per positive cell box/cls losses -----------------
__global__ __launch_bounds__(256)
void k_pos(const float* __restrict__ o0, const float* __restrict__ o1,
           const float* __restrict__ o2, u32* __restrict__ ws) {
    const u32 wgl  = (u32)blockIdx.x * 8u + (u32)(threadIdx.x >> 5);  // one wave / entry
    const int lane = threadIdx.x & 31;
    const u32 list = wgl >> 9;
    const u32 e    = wgl & 511u;
    const int s = (int)(list >> 4), b = (int)(list & 15u);
    const float* base = (s == 0) ? o0 : ((s == 1) ? o1 : o2);
    const int HW = (s == 0) ? 6400 : ((s == 1) ? 1600 : 400);
    const u32 count = ws[COUNTS_U + list];
    float* res = ((float*)ws) + RES_U + (size_t)wgl * 4u;

    float r0 = 0.f, r1 = 0.f, r2 = 0.f, r3 = 0.f;
    if (e < count) {
        const u32* ent = ws + LISTS_U + list * 5120u + e * 10u;
        const float* ef = (const float*)ent;
        const u32 cell = ent[0];
        const float objt = ef[1], w = ef[2];
        const float bx1 = ef[3], by1 = ef[4], bx2 = ef[5], by2 = ef[6];
        const u32 m0 = ent[7], m1 = ent[8], m2 = ent[9];
        const size_t eb = ((size_t)(b * 3) * (size_t)HW + cell) * 85u; // anchor 0
        const bool pos = objt > 0.5f;

        float cv = 0.f;
        if (pos) {
            for (int c = lane; c < NCLS; c += 32) {
                float x = base[eb + 5 + c];
                u32 bit = (((c < 32) ? m0 : ((c < 64) ? m1 : m2)) >> (c & 31)) & 1u;
                cv += softplusf(x) - x * (float)bit;
            }
        }
        for (int off = 16; off; off >>= 1) cv += __shfl_down(cv, off, 32);

        if (lane == 0) {
            float xo = base[eb + 4];
            r0 = xo * objt;                         // obj correction term
            if (pos) {
                float bt[4] = {bx1, by1, bx2, by2};
                float pb[4]; float l1 = 0.f;
                for (int k = 0; k < 4; ++k) {
                    float x = base[eb + k];
                    pb[k] = 1.f / (1.f + expf(-x));
                    float d = pb[k] - bt[k], ad = fabsf(d);
                    l1 += (ad < 1.f) ? (0.5f * d * d) : (ad - 0.5f);
                }
                l1 *= 0.25f;
                float ix1 = fmaxf(pb[0], bx1), iy1 = fmaxf(pb[1], by1);
                float ix2 = fminf(pb[2], bx2), iy2 = fminf(pb[3], by2);
                float inter = fmaxf(ix2 - ix1, 0.f) * fmaxf(iy2 - iy1, 0.f);
                float a1 = fmaxf(pb[2] - pb[0], 0.f) * fmaxf(pb[3] - pb[1], 0.f);
                float a2 = fmaxf(bx2 - bx1, 0.f) * fmaxf(by2 - by1, 0.f);
                float iou = inter / (a1 + a2 - inter + 1e-6f);
                r1 = (l1 + 1.f - iou) * w;
                r2 = (cv / (float)NCLS) * w;
                r3 = 1.f;
            }
        }
    }
    if (lane == 0) { res[0] = r0; res[1] = r1; res[2] = r2; res[3] = r3; }
}

// ---------------- kernel 4: deterministic final combine ----------------------
__device__ __forceinline__ float block_reduce(float v, float* sbuf) {
    int t = threadIdx.x;
    sbuf[t] = v; __syncthreads();
    for (int s = 128; s > 0; s >>= 1) {
        if (t < s) sbuf[t] += sbuf[t + s];
        __syncthreads();
    }
    float r = sbuf[0]; __syncthreads();
    return r;
}

__global__ __launch_bounds__(256)
void k_final(u32* __restrict__ ws, float* __restrict__ out) {
    __shared__ float red[256];
    const float* resf = ((const float*)ws) + RES_U;
    const float* part = ((const float*)ws) + PART_U;
    const int pbeg[4] = {0, NBLK0, NBLK0 + NBLK1, NPART};
    const float ncell[3] = {307200.f, 76800.f, 19200.f};
    float blst[3], olst[3], clst[3];
    for (int s = 0; s < 3; ++s) {
        float oc = 0.f, bx = 0.f, cv = 0.f, np = 0.f, osp = 0.f;
        for (u32 i = threadIdx.x; i < 8192u; i += 256u) {
            const float* r = resf + ((size_t)s * 8192u + i) * 4u;
            oc += r[0]; bx += r[1]; cv += r[2]; np += r[3];
        }
        for (int i = pbeg[s] + threadIdx.x; i < pbeg[s + 1]; i += 256) osp += part[i];
        oc  = block_reduce(oc, red);
        bx  = block_reduce(bx, red);
        cv  = block_reduce(cv, red);
        np  = block_reduce(np, red);
        osp = block_reduce(osp, red);
        float obj_loss = (osp - oc) / ncell[s];
        float den = fmaxf(np, 1.f);
        blst[s] = (np > 0.f) ? bx / den : 0.f;
        olst[s] = obj_loss;
        clst[s] = (np > 0.f) ? cv / den : 0.f;
    }
    if (threadIdx.x == 0) {
        float bl = (blst[0] + blst[1] + blst[2]) / 3.f;
        float ol = (olst[0] + olst[1] + olst[2]) / 3.f;
        float cl = (clst[0] + clst[1] + clst[2]) / 3.f;
        out[0] = 5.f * bl + ol + cl;
        out[1] = bl; out[2] = ol; out[3] = cl;
    }
}

// ---------------- launcher ---------------------------------------------------
extern "C" void kernel_launch(void* const* d_in, const int* in_sizes, int n_in,
                              void* d_out, int out_size, void* d_ws, size_t ws_size,
                              hipStream_t stream) {
    (void)in_sizes; (void)n_in; (void)out_size; (void)ws_size;
    const float* o0  = (const float*)d_in[0];
    const float* o1  = (const float*)d_in[1];
    const float* o2  = (const float*)d_in[2];
    const float* tgt = (const float*)d_in[3];
    u32* ws = (u32*)d_ws;
    float* out = (float*)d_out;

    k_build<<<48, 64, 0, stream>>>(tgt, ws);
    k_obj<<<NPART, 256, 0, stream>>>(o0, o1, o2, ws);
    k_pos<<<3072, 256, 0, stream>>>(o0, o1, o2, ws);
    k_final<<<1, 256, 0, stream>>>(ws, out);
}